// SparseResCoCNModuleN_76459007803904
// MI455X (gfx1250) — compile-verified
//
#include <hip/hip_runtime.h>
#include <math.h>

typedef float v2f __attribute__((ext_vector_type(2)));
typedef float v8f __attribute__((ext_vector_type(8)));

#define NROWS  50000
#define DIN    128
#define DMODEL 256
#define HH     4
#define DCAT   1024          // HH * DMODEL
#define NCLS   40
#define NCLSP  48            // padded head columns (3 x 16)
#define NTILES 3125          // NROWS / 16

// workspace layout (float offsets)
#define OFF_G    0ull            // NROWS*DMODEL = 12,800,000
#define OFF_W    12800000ull     // HH*NROWS    =    200,000
#define OFF_MU   13000000ull     // NROWS
#define OFF_RSTD 13050000ull     // NROWS
#define OFF_U    13100000ull     // 40
#define OFF_T    13100040ull     // 40
#define OFF_WT   13110000ull     // W_in^T: 256*256 = 65,536
#define OFF_WHT  13180000ull     // W_head^T padded: 48*1024 = 49,152
// total ~13,229,152 floats = 52.9 MB

static __device__ __forceinline__ v8f wmma4(v2f a, v2f b, v8f c) {
    // D = A(16x4,f32) * B(4x16,f32) + C(16x16,f32), exact fp32 via CDNA5 WMMA
    return __builtin_amdgcn_wmma_f32_16x16x4_f32(false, a, false, b, (short)0, c,
                                                 false, false);
}

// ---------------------------------------------------------------------------
// 0a) transpose W_in -> Wt[col*256 + k]
// ---------------------------------------------------------------------------
__global__ void k_tr_in(const float* __restrict__ W_in, float* __restrict__ Wt) {
    int idx = blockIdx.x * blockDim.x + threadIdx.x;   // 65536 threads
    int col = idx >> 8;
    int k   = idx & 255;
    Wt[col * DMODEL + k] = W_in[k * DMODEL + col];
}

// ---------------------------------------------------------------------------
// 0b) transpose + pad W_head -> Wht[col*1024 + k], cols 40..47 zero
// ---------------------------------------------------------------------------
__global__ void k_tr_head(const float* __restrict__ W_head, float* __restrict__ Wht) {
    int idx = blockIdx.x * blockDim.x + threadIdx.x;   // 49152 threads
    int col = idx >> 10;
    int k   = idx & 1023;
    Wht[idx] = (col < NCLS) ? W_head[k * NCLS + col] : 0.f;
}

// ---------------------------------------------------------------------------
// 1) scatter permutation weights: w[b][j] = v for each nonzero (b, i, j)
// ---------------------------------------------------------------------------
__global__ void k_perm(const int* __restrict__ idx, const float* __restrict__ vals,
                       float* __restrict__ w) {
    const int total = HH * NROWS;
    int k = blockIdx.x * blockDim.x + threadIdx.x;
    if (k >= total) return;
    int b = idx[k];               // indices[0]
    int j = idx[2 * total + k];   // indices[2] = sigma_b(i)
    w[b * NROWS + j] = vals[k];   // each (b,j) hit exactly once (permutation)
}

// ---------------------------------------------------------------------------
// 2) G = concat(features, appd) @ W_in   (NROWS x 256) @ (256 x 256)
//    wave = one 16x64 output tile (4 WMMA f32 accumulators), K stepped by 4
//    B-fragments come from transposed weights -> single b64 load per tile/step
// ---------------------------------------------------------------------------
__global__ void k_gemm_in(const float* __restrict__ features,
                          const float* __restrict__ appd,
                          const float* __restrict__ Wt,   // [col*256 + k]
                          float* __restrict__ G) {
    const int tid  = threadIdx.x;
    const int wave = tid >> 5;        // 0..7
    const int lane = tid & 31;
    const int half = lane >> 4;       // 0/1
    const int mrow = lane & 15;

    const int tile = blockIdx.x * 2 + (wave >> 2);   // wave-uniform
    if (tile >= NTILES) return;                      // whole-wave exit (EXEC stays full)
    const int cbase = (wave & 3) * 64;
    const int row   = tile * 16 + mrow;

    v8f acc[4] = {v8f{0}, v8f{0}, v8f{0}, v8f{0}};

    const float* __restrict__ arow = features + row * DIN;
#pragma unroll 2
    for (int k0 = 0; k0 < DIN; k0 += 4) {
        const int ka = k0 + 2 * half;
        v2f a;
        a.x = arow[ka];
        a.y = arow[ka + 1];
#pragma unroll
        for (int t = 0; t < 4; ++t) {
            const int col = cbase + t * 16 + mrow;
            v2f b;
            b.x = Wt[col * DMODEL + ka];
            b.y = Wt[col * DMODEL + ka + 1];
            acc[t] = wmma4(a, b, acc[t]);
        }
    }
    const float* __restrict__ arow2 = appd + row * DIN;
#pragma unroll 2
    for (int k0 = 0; k0 < DIN; k0 += 4) {
        const int ka = k0 + 2 * half;
        v2f a;
        a.x = arow2[ka];
        a.y = arow2[ka + 1];
#pragma unroll
        for (int t = 0; t < 4; ++t) {
            const int col = cbase + t * 16 + mrow;
            v2f b;
            b.x = Wt[col * DMODEL + (DIN + ka)];
            b.y = Wt[col * DMODEL + (DIN + ka + 1)];
            acc[t] = wmma4(a, b, acc[t]);
        }
    }
#pragma unroll
    for (int r = 0; r < 8; ++r) {
        const int orow = tile * 16 + half * 8 + r;   // C/D: VGPR r -> M = r + 8*half
#pragma unroll
        for (int t = 0; t < 4; ++t)
            G[orow * DMODEL + cbase + t * 16 + mrow] = acc[t][r];
    }
}

// ---------------------------------------------------------------------------
// 3) u[o] = sum_k gamma[k]*W_head[k,o] ;  t[o] = sum_k beta[k]*W_head[k,o] + b_head[o]
// ---------------------------------------------------------------------------
__global__ void k_ut(const float* __restrict__ gamma, const float* __restrict__ beta,
                     const float* __restrict__ W_head, const float* __restrict__ b_head,
                     float* __restrict__ u, float* __restrict__ t) {
    int o = threadIdx.x;
    if (o >= NCLS) return;
    float su = 0.f, st = 0.f;
    for (int k = 0; k < DCAT; ++k) {
        float wv = W_head[k * NCLS + o];
        su += gamma[k] * wv;
        st += beta[k] * wv;
    }
    u[o] = su;
    t[o] = st + b_head[o];
}

// ---------------------------------------------------------------------------
// 4) per-row LN stats over y[b,j,c] = w*relu(w*G + b_in); one wave per row
// ---------------------------------------------------------------------------
__global__ void k_stats(const float* __restrict__ G, const float* __restrict__ w,
                        const float* __restrict__ b_in,
                        float* __restrict__ mu, float* __restrict__ rstd) {
    const int wave = threadIdx.x >> 5;
    const int lane = threadIdx.x & 31;
    const int j = blockIdx.x * 8 + wave;
    if (j >= NROWS) return;

    float wb[HH];
#pragma unroll
    for (int b = 0; b < HH; ++b) wb[b] = w[b * NROWS + j];

    float s = 0.f, ss = 0.f;
#pragma unroll
    for (int c = lane; c < DMODEL; c += 32) {
        const float g  = G[j * DMODEL + c];
        const float bi = b_in[c];
#pragma unroll
        for (int b = 0; b < HH; ++b) {
            const float y = wb[b] * fmaxf(wb[b] * g + bi, 0.f);
            s += y;
            ss += y * y;
        }
    }
#pragma unroll
    for (int m = 16; m >= 1; m >>= 1) {
        s  += __shfl_xor(s, m, 32);
        ss += __shfl_xor(ss, m, 32);
    }
    if (lane == 0) {
        const float mean = s * (1.0f / DCAT);
        const float var  = ss * (1.0f / DCAT) - mean * mean;
        mu[j]   = mean;
        rstd[j] = rsqrtf(var + 1e-5f);
    }
}

// ---------------------------------------------------------------------------
// 5) out = rstd*( (y*gamma) @ W_head - mu*u ) + t ; wave = 16 rows x 48 cols
//    A-fragments of z = gamma * w * relu(w*G + b_in) built on the fly;
//    B-fragments from transposed/padded W_head (b64 loads, no masking)
// ---------------------------------------------------------------------------
__global__ void k_head(const float* __restrict__ G, const float* __restrict__ w,
                       const float* __restrict__ b_in, const float* __restrict__ gamma,
                       const float* __restrict__ Wht,   // [col*1024 + k], 48 cols
                       const float* __restrict__ mu, const float* __restrict__ rstd,
                       const float* __restrict__ u, const float* __restrict__ tvec,
                       float* __restrict__ out) {
    const int tid  = threadIdx.x;
    const int wave = tid >> 5;
    const int lane = tid & 31;
    const int half = lane >> 4;
    const int mrow = lane & 15;

    const int tile = blockIdx.x * 8 + wave;          // wave-uniform
    if (tile >= NTILES) return;
    const int row = tile * 16 + mrow;

    float wb[HH];
#pragma unroll
    for (int b = 0; b < HH; ++b) wb[b] = w[b * NROWS + row];

    v8f acc[3] = {v8f{0}, v8f{0}, v8f{0}};

#pragma unroll 2
    for (int k0 = 0; k0 < DCAT; k0 += 4) {
        const int ka = k0 + 2 * half;                // even; ka,ka+1 share b = k0>>8
        const int b  = k0 >> 8;
        const int c  = ka & (DMODEL - 1);
        const float wv = wb[b];
        const float g0 = G[row * DMODEL + c];
        const float g1 = G[row * DMODEL + c + 1];
        v2f a;
        a.x = gamma[ka]     * wv * fmaxf(wv * g0 + b_in[c],     0.f);
        a.y = gamma[ka + 1] * wv * fmaxf(wv * g1 + b_in[c + 1], 0.f);
#pragma unroll
        for (int t = 0; t < 3; ++t) {
            const int col = t * 16 + mrow;           // 0..47 (cols >=40 are zero in Wht)
            v2f bf;
            bf.x = Wht[col * DCAT + ka];
            bf.y = Wht[col * DCAT + ka + 1];
            acc[t] = wmma4(a, bf, acc[t]);
        }
    }
#pragma unroll
    for (int r = 0; r < 8; ++r) {
        const int orow = tile * 16 + half * 8 + r;
        const float rs = rstd[orow];
        const float m_ = mu[orow];
#pragma unroll
        for (int t = 0; t < 3; ++t) {
            const int col = t * 16 + mrow;
            if (col < NCLS)
                out[orow * NCLS + col] = rs * (acc[t][r] - m_ * u[col]) + tvec[col];
        }
    }
}

// ---------------------------------------------------------------------------
extern "C" void kernel_launch(void* const* d_in, const int* in_sizes, int n_in,
                              void* d_out, int out_size, void* d_ws, size_t ws_size,
                              hipStream_t stream) {
    const int*   perm_idx  = (const int*)  d_in[0];
    const float* perm_vals = (const float*)d_in[1];
    const float* features  = (const float*)d_in[2];
    const float* appd      = (const float*)d_in[3];
    const float* W_in      = (const float*)d_in[4];
    const float* b_in      = (const float*)d_in[5];
    const float* ln_gamma  = (const float*)d_in[6];
    const float* ln_beta   = (const float*)d_in[7];
    const float* W_head    = (const float*)d_in[8];
    const float* b_head    = (const float*)d_in[9];
    float* out = (float*)d_out;

    float* ws   = (float*)d_ws;
    float* G    = ws + OFF_G;
    float* w    = ws + OFF_W;
    float* mu   = ws + OFF_MU;
    float* rstd = ws + OFF_RSTD;
    float* u    = ws + OFF_U;
    float* tv   = ws + OFF_T;
    float* Wt   = ws + OFF_WT;
    float* Wht  = ws + OFF_WHT;

    // 0) stage transposed weights (tiny; caches keep them hot for the GEMMs)
    k_tr_in<<<(DMODEL * DMODEL) / 256, 256, 0, stream>>>(W_in, Wt);
    k_tr_head<<<(NCLSP * DCAT) / 256, 256, 0, stream>>>(W_head, Wht);
    // 1) permutation weight scatter
    k_perm<<<(HH * NROWS + 255) / 256, 256, 0, stream>>>(perm_idx, perm_vals, w);
    // 2) shared input GEMM (WMMA f32): 2 row-tiles per 256-thread block
    k_gemm_in<<<(NTILES + 1) / 2, 256, 0, stream>>>(features, appd, Wt, G);
    // 3) LN-fold vectors
    k_ut<<<1, 64, 0, stream>>>(ln_gamma, ln_beta, W_head, b_head, u, tv);
    // 4) per-row LN statistics
    k_stats<<<(NROWS + 7) / 8, 256, 0, stream>>>(G, w, b_in, mu, rstd);
    // 5) head GEMM (WMMA f32) with fused scale/relu/gamma and LN fixup epilogue
    k_head<<<(NTILES + 7) / 8, 256, 0, stream>>>(G, w, b_in, ln_gamma, Wht,
                                                 mu, rstd, u, tv, out);
}